// SparseGATLayer_14276471292068
// MI455X (gfx1250) — compile-verified
//
#include <hip/hip_runtime.h>
#include <math.h>

#define N_NODES 50000
#define N_EDGES 800000
#define IN_DIM  256
#define OUT_DIM 64
#define ALPHA   0.2f
#define GAT_EPS 9e-15f
#define M_TILES (N_NODES / 16)   // 3125, exact

typedef float v2f __attribute__((ext_vector_type(2)));
typedef float v8f __attribute__((ext_vector_type(8)));

// Monotonic float<->uint mapping so unsigned atomicMax == float max.
__device__ __forceinline__ unsigned enc_f32(float f) {
    unsigned u = __float_as_uint(f);
    return (u & 0x80000000u) ? ~u : (u | 0x80000000u);
}
__device__ __forceinline__ float dec_f32(unsigned e) {
    unsigned u = (e & 0x80000000u) ? (e ^ 0x80000000u) : ~e;
    return __uint_as_float(u);
}

// ---------------- zero init (out, rowsum, maxslot) ----------------
__global__ void zero_kernel(float* __restrict__ out, float* __restrict__ rowsum,
                            unsigned* __restrict__ maxslot) {
    long i = (long)blockIdx.x * blockDim.x + threadIdx.x;
    long tot = (long)N_NODES * OUT_DIM;
    if (i < tot) out[i] = 0.0f;
    if (i < N_NODES) rowsum[i] = 0.0f;
    if (i == 0) *maxslot = 0u;   // 0 < enc(-inf): safe identity for encoded max
}

// ---------------- h = x @ W via V_WMMA_F32_16X16X4_F32 ----------------
// One wave -> one 16x16 tile of h. Block = 8 waves = 2 m-tiles x 4 n-tiles.
__global__ void gemm_wmma_f32(const float* __restrict__ x,
                              const float* __restrict__ W,
                              float* __restrict__ h) {
    const int lane = threadIdx.x & 31;
    const int wave = threadIdx.x >> 5;
    const int mt   = blockIdx.x * 2 + (wave >> 2);
    if (mt >= M_TILES) return;                    // wave-uniform; EXEC stays all-1s
    const int n0   = (wave & 3) * 16;
    const int mrow = (lane & 15);                 // M index within tile for A
    const int koff = (lane >> 4) * 2;             // K sub-offset: lanes 16-31 hold K+2,K+3
    const int col  = n0 + (lane & 15);            // N index for B/C/D

    const float* xr = x + (long)(mt * 16 + mrow) * IN_DIM + koff;
    v8f c = {0.f, 0.f, 0.f, 0.f, 0.f, 0.f, 0.f, 0.f};

#pragma unroll 4
    for (int k = 0; k < IN_DIM; k += 4) {
        // A fragment: VGPR0=A[m][k+koff], VGPR1=A[m][k+koff+1] (contiguous pair)
        v2f a = *(const v2f*)(xr + k);
        // B fragment: VGPR v holds row K = k+koff+v (lanes 0-15: K=k..; 16-31: K=k+2..)
        v2f b;
        b.x = W[(long)(k + koff)     * OUT_DIM + col];
        b.y = W[(long)(k + koff + 1) * OUT_DIM + col];
        c = __builtin_amdgcn_wmma_f32_16x16x4_f32(
                /*neg_a=*/false, a, /*neg_b=*/false, b,
                /*c_mod=*/(short)0, c, /*reuse_a=*/false, /*reuse_b=*/false);
    }

    // D layout: VGPR v -> M = v + 8*(lane>=16), N = lane&15
    float* hp = h + (long)(mt * 16 + (lane >> 4) * 8) * OUT_DIM + col;
#pragma unroll
    for (int v = 0; v < 8; ++v) hp[(long)v * OUT_DIM] = c[v];
}

// ---------------- s = h @ a_src, t = h @ a_dst ----------------
__global__ void st_kernel(const float* __restrict__ h, const float* __restrict__ attn,
                          float* __restrict__ s, float* __restrict__ t) {
    int i = blockIdx.x * blockDim.x + threadIdx.x;
    if (i >= N_NODES) return;
    const float* hr = h + (long)i * OUT_DIM;
    float ss = 0.f, tt = 0.f;
#pragma unroll 8
    for (int d = 0; d < OUT_DIM; ++d) {
        float hv = hr[d];
        ss += hv * attn[d];
        tt += hv * attn[OUT_DIM + d];
    }
    s[i] = ss;
    t[i] = tt;
}

// ---------------- global max of leaky-relu edge scores ----------------
__global__ void edge_max_kernel(const int* __restrict__ edge,
                                const float* __restrict__ s, const float* __restrict__ t,
                                unsigned* __restrict__ maxslot) {
    __shared__ unsigned red[256];
    unsigned local = 0u;
    for (int e = blockIdx.x * blockDim.x + threadIdx.x; e < N_EDGES;
         e += gridDim.x * blockDim.x) {
        int src = edge[e];
        int dst = edge[N_EDGES + e];
        float v = s[src] + t[dst];
        v = (v >= 0.f) ? v : ALPHA * v;
        unsigned enc = enc_f32(v);
        local = (enc > local) ? enc : local;
    }
    red[threadIdx.x] = local;
    __syncthreads();
    for (int off = 128; off > 0; off >>= 1) {
        if ((int)threadIdx.x < off) {
            unsigned o = red[threadIdx.x + off];
            if (o > red[threadIdx.x]) red[threadIdx.x] = o;
        }
        __syncthreads();
    }
    if (threadIdx.x == 0) atomicMax(maxslot, red[0]);
}

// ---------------- edge_e = exp(score - max); rowsum[src] += edge_e ----------------
__global__ void edge_e_kernel(const int* __restrict__ edge,
                              const float* __restrict__ s, const float* __restrict__ t,
                              const unsigned* __restrict__ maxslot,
                              float* __restrict__ edge_e, float* __restrict__ rowsum) {
    float vmax = dec_f32(*maxslot);
    for (int e = blockIdx.x * blockDim.x + threadIdx.x; e < N_EDGES;
         e += gridDim.x * blockDim.x) {
        int src = edge[e];
        int dst = edge[N_EDGES + e];
        float v = s[src] + t[dst];
        v = (v >= 0.f) ? v : ALPHA * v;
        float ee = expf(v - vmax);
        edge_e[e] = ee;
        atomicAdd(&rowsum[src], ee);
    }
}

// ---------------- out[src] += edge_e * h[dst] (wave per edge, lane owns 2 dims) ----
__global__ void scatter_kernel(const int* __restrict__ edge,
                               const float* __restrict__ edge_e,
                               const float* __restrict__ h,
                               float* __restrict__ out) {
    const int lane = threadIdx.x & 31;
    const int wid  = (blockIdx.x * blockDim.x + threadIdx.x) >> 5;
    const int nw   = (gridDim.x * blockDim.x) >> 5;
    for (int e = wid; e < N_EDGES; e += nw) {
        int src = edge[e];
        int dst = edge[N_EDGES + e];
        float ee = edge_e[e];
        const float* hd = h + (long)dst * OUT_DIM;
        float* op = out + (long)src * OUT_DIM;
        atomicAdd(&op[lane],      ee * hd[lane]);
        atomicAdd(&op[lane + 32], ee * hd[lane + 32]);
    }
}

// ---------------- out /= (rowsum + eps) ----------------
__global__ void norm_kernel(float* __restrict__ out, const float* __restrict__ rowsum) {
    long i = (long)blockIdx.x * blockDim.x + threadIdx.x;
    if (i >= (long)N_NODES * OUT_DIM) return;
    out[i] = out[i] / (rowsum[i >> 6] + GAT_EPS);
}

extern "C" void kernel_launch(void* const* d_in, const int* in_sizes, int n_in,
                              void* d_out, int out_size, void* d_ws, size_t ws_size,
                              hipStream_t stream) {
    const float* x    = (const float*)d_in[0];   // N x 256
    const int*   edge = (const int*)d_in[1];     // 2 x E (row 0 = src, row 1 = dst)
    const float* W    = (const float*)d_in[2];   // 256 x 64
    const float* attn = (const float*)d_in[3];   // 1 x 128
    float* out = (float*)d_out;                  // N x 64

    // workspace partition (floats)
    float* ws      = (float*)d_ws;
    float* h       = ws;                                  // N*64
    float* s       = h + (long)N_NODES * OUT_DIM;         // N
    float* t       = s + N_NODES;                         // N
    float* rowsum  = t + N_NODES;                         // N
    float* edge_e  = rowsum + N_NODES;                    // E
    unsigned* maxslot = (unsigned*)(edge_e + N_EDGES);    // 1

    const long n_out_elems = (long)N_NODES * OUT_DIM;     // 3.2M

    zero_kernel<<<(int)((n_out_elems + 255) / 256), 256, 0, stream>>>(out, rowsum, maxslot);

    // GEMM: 2 m-tiles per block, 3125 m-tiles total
    gemm_wmma_f32<<<(M_TILES + 1) / 2, 256, 0, stream>>>(x, W, h);

    st_kernel<<<(N_NODES + 255) / 256, 256, 0, stream>>>(h, attn, s, t);

    edge_max_kernel<<<1024, 256, 0, stream>>>(edge, s, t, maxslot);

    edge_e_kernel<<<2048, 256, 0, stream>>>(edge, s, t, maxslot, edge_e, rowsum);

    // 25000 blocks * 8 waves = 200000 waves, 4 edges per wave
    scatter_kernel<<<25000, 256, 0, stream>>>(edge, edge_e, h, out);

    norm_kernel<<<(int)((n_out_elems + 255) / 256), 256, 0, stream>>>(out, rowsum);
}